// MessagePassing_67886253080943
// MI455X (gfx1250) — compile-verified
//
#include <hip/hip_runtime.h>

// out[dst] += x[src] per edge; N=50000, E=800000, D=64, fp32.
// L2-resident scatter-add (x=12.8MB, out=12.8MB << 192MB L2).
// Hot path: 16 lanes/edge, b128 row gather + no-return global_atomic_add_f32.
// CDNA5 path: edge indices staged to LDS via global_load_async_to_lds_b128
// (ASYNCcnt), double-buffered so the index DMA overlaps the atomics.

#define NODE_D 64
#define CHUNK 256              // edges per LDS chunk
#define PASSES (CHUNK / 16)    // 16 edges (one per half-wave group) per pass

__global__ __launch_bounds__(256) void mp_zero_out(float4* __restrict__ out4, int n4) {
    int i = blockIdx.x * blockDim.x + threadIdx.x;
    const int stride = gridDim.x * blockDim.x;
    const float4 z = make_float4(0.f, 0.f, 0.f, 0.f);
    for (; i < n4; i += stride) out4[i] = z;
}

__global__ __launch_bounds__(256) void mp_scatter_async(
    const int* __restrict__ edge_dst,
    const int* __restrict__ edge_src,
    const float* __restrict__ x,
    float* __restrict__ out,
    int E) {
    // [buffer][0=dst,1=src][edge-in-chunk]
    __shared__ __align__(16) int sbuf[2][2][CHUNK];

    const int t    = threadIdx.x;
    const int lane = t & 15;      // element group within the 256B row
    const int grp  = t >> 4;      // 0..15: half-wave group -> edge within pass

    const int nchunks = E / CHUNK;
    const int cstride = gridDim.x;
    int c = blockIdx.x;

    // Issue one async 16B copy per lane: waves 0-1 stage dst, waves 2-3 stage src.
    auto issue_chunk = [&](int chunk, int b) {
        if (t < 128) {
            const int half = t >> 6;   // 0 = dst row, 1 = src row
            const int li   = t & 63;   // 64 lanes x 4 ints = 256 indices
            const int* g   = (half ? edge_src : edge_dst) + chunk * CHUNK + li * 4;
            const unsigned loff =
                (unsigned)(size_t)(&sbuf[b][half][li * 4]);   // low 32 bits = LDS offset
            asm volatile("global_load_async_to_lds_b128 %0, %1, off"
                         :
                         : "v"(loff), "v"(g)
                         : "memory");
        }
    };

    if (c < nchunks) issue_chunk(c, 0);
    int buf = 0;

    for (; c < nchunks; c += cstride, buf ^= 1) {
        const int cn = c + cstride;
        if (cn < nchunks) {
            issue_chunk(cn, buf ^ 1);                     // DMA next chunk ahead
            asm volatile("s_wait_asynccnt 0x1" ::: "memory"); // current chunk done
        } else {
            asm volatile("s_wait_asynccnt 0x0" ::: "memory");
        }
        __syncthreads();   // publish LDS indices to all 8 waves

        #pragma unroll 4
        for (int p = 0; p < PASSES; ++p) {
            const int el  = p * 16 + grp;
            const int dst = sbuf[buf][0][el];
            const int src = sbuf[buf][1][el];

            // Prefetch the next pass's source row while this edge's work issues.
            if (p + 1 < PASSES) {
                const int s2 = sbuf[buf][1][(p + 1) * 16 + grp];
                __builtin_prefetch(x + (size_t)s2 * NODE_D + lane * 4, 0, 3);
            }

            const float4 v = *reinterpret_cast<const float4*>(
                x + (size_t)src * NODE_D + (size_t)lane * 4);
            float* o = out + (size_t)dst * NODE_D + (size_t)lane * 4;
            atomicAdd(o + 0, v.x);
            atomicAdd(o + 1, v.y);
            atomicAdd(o + 2, v.z);
            atomicAdd(o + 3, v.w);
        }
        __syncthreads();   // all reads done before this buffer is re-targeted
    }

    // Tail (E % CHUNK edges): direct loads, block 0 only.
    if (blockIdx.x == 0) {
        const int tail_base = nchunks * CHUNK;
        for (int i = grp; i < E - tail_base; i += 16) {
            const int e   = tail_base + i;
            const int dst = edge_dst[e];
            const int src = edge_src[e];
            const float4 v = *reinterpret_cast<const float4*>(
                x + (size_t)src * NODE_D + (size_t)lane * 4);
            float* o = out + (size_t)dst * NODE_D + (size_t)lane * 4;
            atomicAdd(o + 0, v.x);
            atomicAdd(o + 1, v.y);
            atomicAdd(o + 2, v.z);
            atomicAdd(o + 3, v.w);
        }
    }
}

extern "C" void kernel_launch(void* const* d_in, const int* in_sizes, int n_in,
                              void* d_out, int out_size, void* d_ws, size_t ws_size,
                              hipStream_t stream) {
    (void)n_in; (void)d_ws; (void)ws_size;

    const int* edge_index = (const int*)d_in[0];   // [2, E] flat int32
    const float* x        = (const float*)d_in[1]; // [N, 64] fp32
    float* out            = (float*)d_out;         // [N, 64] fp32

    const int E = in_sizes[0] / 2;
    const int* edge_dst = edge_index;       // row 0: targets
    const int* edge_src = edge_index + E;   // row 1: sources

    // 1) Zero the output (harness poisons d_out).
    {
        const int n4 = out_size / 4;
        const int threads = 256;
        int blocks = (n4 + threads - 1) / threads;
        if (blocks > 4096) blocks = 4096;
        mp_zero_out<<<blocks, threads, 0, stream>>>((float4*)out, n4);
    }

    // 2) Async-staged scatter-add: ~3 chunks per block so the DMA pipeline
    //    actually overlaps; 8 wave32 per block.
    {
        const int nchunks = E / CHUNK;
        int blocks = nchunks < 1024 ? (nchunks > 0 ? nchunks : 1) : 1024;
        mp_scatter_async<<<blocks, 256, 0, stream>>>(edge_dst, edge_src, x, out, E);
    }
}